// ShotNoise_32040456028521
// MI455X (gfx1250) — compile-verified
//
#include <hip/hip_runtime.h>

// ShotNoise: out = Poisson(img * 12) / 12, fp32 in / fp32 out.
// Memory-bound floor ~17us @ 23.3 TB/s; Poisson inversion makes it mildly
// compute-bound, so: cheap counter-based RNG, merged 4-wide inversion loop,
// B128 async global->LDS staging (ASYNCcnt pipeline), NT stores.

typedef __attribute__((ext_vector_type(4))) float v4f;

#define BLOCK 256
#define LAM_COEF 12.0f
#define INV_LAM (1.0f / 12.0f)
#define MAX_K 64

// ---- deterministic counter-based uniform in [0,1) (splitmix64 finalizer) ----
__device__ __forceinline__ float u01(unsigned long long x) {
  x ^= 0x9E3779B97F4A7C15ull;            // fixed seed/stream
  x ^= x >> 33; x *= 0xFF51AFD7ED558CCDull;
  x ^= x >> 33; x *= 0xC4CEB9FE1A85EC53ull;
  x ^= x >> 33;
  return (float)(unsigned int)(x >> 40) * 0x1.0p-24f;   // 24-bit mantissa
}

// ---- Poisson(lam)/12 via inversion-by-sequential-search, scalar path ----
__device__ __forceinline__ float poisson12_scalar(float img, unsigned long long idx) {
  float lam = img * LAM_COEF;
  float u = u01(idx);
  float p = __expf(-lam);
  float F = p;
  float k = 0.0f;
#pragma unroll 1
  for (int it = 1; it <= MAX_K; ++it) {
    if (!(u > F)) break;
    float r = __builtin_amdgcn_rcpf((float)it);
    p = p * lam * r;
    F += p;
    k += 1.0f;
  }
  return k * INV_LAM;
}

// ---- async global -> LDS copy of one 16B chunk per lane (gfx1250 TDM-lite) ----
__device__ __forceinline__ void async_copy_b128(unsigned lds_byte_addr, const v4f* gaddr) {
  asm volatile("global_load_async_to_lds_b128 %0, %1, off"
               :: "v"(lds_byte_addr), "v"(gaddr)
               : "memory");
}
__device__ __forceinline__ void wait_asynccnt_le1() {
  asm volatile("s_wait_asynccnt 1" ::: "memory");
}
__device__ __forceinline__ void wait_asynccnt_0() {
  asm volatile("s_wait_asynccnt 0" ::: "memory");
}

__global__ __launch_bounds__(BLOCK)
void shotnoise_kernel(const float* __restrict__ in, float* __restrict__ out, long long n) {
  __shared__ v4f buf[2][BLOCK];

  const int tid = threadIdx.x;
  const long long n4 = n >> 2;                 // float4 count
  const long long ntiles = n4 / BLOCK;         // full tiles of 256 float4s
  const long long tstride = gridDim.x;

  const v4f* __restrict__ in4 = (const v4f*)in;
  v4f* __restrict__ out4 = (v4f*)out;

  // Wave-relative LDS byte addresses of this lane's slot in each buffer.
  // Generic shared pointers carry the LDS offset in their low 32 bits.
  const unsigned lds0 = (unsigned)(unsigned long long)&buf[0][tid];
  const unsigned lds1 = (unsigned)(unsigned long long)&buf[1][tid];

  long long t = blockIdx.x;
  int cur = 0;

  // Prologue: kick off the first tile's async copy.
  if (t < ntiles) {
    async_copy_b128(lds0, in4 + t * BLOCK + tid);
  }

#pragma unroll 1
  for (; t < ntiles; t += tstride) {
    const long long tn = t + tstride;
    if (tn < ntiles) {
      // Issue next tile into the other buffer, then wait until only that
      // newest copy is outstanding (=> current tile's data is in LDS).
      async_copy_b128(cur ? lds0 : lds1, in4 + tn * BLOCK + tid);
      wait_asynccnt_le1();
    } else {
      wait_asynccnt_0();
    }

    // Prefetch two tiles ahead into GL2 (speculative, safe if OOB-dropped).
    const long long tp = t + 2 * tstride;
    if (tp < ntiles) {
      __builtin_prefetch((const void*)(in4 + tp * BLOCK + tid), 0, 0);
    }

    // Each lane consumes exactly the 16 bytes it requested: no barrier needed,
    // the ASYNCcnt wait above is the ordering point.
    v4f x = buf[cur][tid];

    const unsigned long long e = ((unsigned long long)(t * BLOCK + tid)) << 2;

    float lamx = x.x * LAM_COEF;
    float lamy = x.y * LAM_COEF;
    float lamz = x.z * LAM_COEF;
    float lamw = x.w * LAM_COEF;

    float ux = u01(e + 0), uy = u01(e + 1), uz = u01(e + 2), uw = u01(e + 3);

    float px = __expf(-lamx), py = __expf(-lamy), pz = __expf(-lamz), pw = __expf(-lamw);
    float Fx = px, Fy = py, Fz = pz, Fw = pw;
    float kx = 0.0f, ky = 0.0f, kz = 0.0f, kw = 0.0f;

    // Merged inversion loop: one rcp(it) shared by all 4 elements; each
    // element's condition is prefix-true, so k_i counts its active iterations.
#pragma unroll 1
    for (int it = 1; it <= MAX_K; ++it) {
      const bool c0 = ux > Fx, c1 = uy > Fy, c2 = uz > Fz, c3 = uw > Fw;
      if (!(c0 || c1 || c2 || c3)) break;
      const float r = __builtin_amdgcn_rcpf((float)it);
      if (c0) { px = px * lamx * r; Fx += px; kx += 1.0f; }
      if (c1) { py = py * lamy * r; Fy += py; ky += 1.0f; }
      if (c2) { pz = pz * lamz * r; Fz += pz; kz += 1.0f; }
      if (c3) { pw = pw * lamw * r; Fw += pw; kw += 1.0f; }
    }

    v4f rvec;
    rvec.x = kx * INV_LAM;
    rvec.y = ky * INV_LAM;
    rvec.z = kz * INV_LAM;
    rvec.w = kw * INV_LAM;

    // Streaming output: never re-read, keep it out of near caches.
    __builtin_nontemporal_store(rvec, out4 + t * BLOCK + tid);

    cur ^= 1;
  }

  // Tail: any elements beyond the last full tile (none for 64x3x512x512).
  const long long tail_base = ntiles * (long long)BLOCK * 4;
  for (long long i = tail_base + (long long)blockIdx.x * BLOCK + tid; i < n;
       i += tstride * BLOCK) {
    out[i] = poisson12_scalar(in[i], (unsigned long long)i);
  }
}

extern "C" void kernel_launch(void* const* d_in, const int* in_sizes, int n_in,
                              void* d_out, int out_size, void* d_ws, size_t ws_size,
                              hipStream_t stream) {
  (void)in_sizes; (void)n_in; (void)d_ws; (void)ws_size;
  const float* in = (const float*)d_in[0];
  float* out = (float*)d_out;
  const long long n = (long long)out_size;

  long long ntiles = (n / 4) / BLOCK;          // 49,152 for the reference shape
  long long grid = ntiles;
  if (grid > 4096) grid = 4096;                // persistent-ish: >=12 tiles/block
  if (grid < 1) grid = 1;

  shotnoise_kernel<<<(int)grid, BLOCK, 0, stream>>>(in, out, n);
}